// RNN_11347303596519
// MI455X (gfx1250) — compile-verified
//
#include <hip/hip_runtime.h>

// GRU scan (seq=2048, batch=1024, H=19) on CDNA5 V_WMMA_F32_16X16X4_F32.
// One wave32 per 16 batch rows (64 blocks x 32 threads).
// - Tiles 0-1 (r/z gates): single merged K=40 WMMA chain over [x ; h] with
//   biases in the trailing K=19 row (A slot holds constant 1.0).
// - Tiles 2-3 (n gates need x/h parts separate): h-chain in-step; the x-chain
//   for step t+1 is computed during step t's elementwise phase (overlaps the
//   trans pipe) and carried in registers.
// - Hidden state h lives permanently in the A-fragment registers.
// - Activations: native gfx1250 V_TANH_F32 (3 trans ops per element).
// - Single-wave blocks: in-order LDS, compiler-only fences, no s_barrier.

typedef float v2f __attribute__((ext_vector_type(2)));
typedef float v8f __attribute__((ext_vector_type(8)));

#define HID   19
#define SEQ   2048
#define BATCH 1024
#define NKIT  5   // K iterations of 4 covering K=19 (+ bias row = 20)

#define WMMA4(a, b, c) __builtin_amdgcn_wmma_f32_16x16x4_f32( \
        false, (a), false, (b), (short)0, (c), false, false)

__device__ __forceinline__ float fast_tanh(float x) {
#if __has_builtin(__builtin_amdgcn_tanhf)
    return __builtin_amdgcn_tanhf(x);
#else
    float r;
    asm("v_tanh_f32 %0, %1" : "=v"(r) : "v"(x));
    return r;
#endif
}
// sigmoid(x) = 0.5 + 0.5*tanh(x/2): single transcendental op.
__device__ __forceinline__ float fast_sigmoid(float x) {
    return __builtin_fmaf(0.5f, fast_tanh(0.5f * x), 0.5f);
}
__device__ __forceinline__ void wave_sync() {
    __builtin_amdgcn_wave_barrier();   // compiler fence; LDS is in-order per wave
}

__global__ __launch_bounds__(32) void gru_wmma_scan(
    const float* __restrict__ x,    // (SEQ, BATCH, HID)
    const float* __restrict__ Wih,  // (57, 19)
    const float* __restrict__ Whh,  // (57, 19)
    const float* __restrict__ bih,  // (57)
    const float* __restrict__ bhh,  // (57)
    const float* __restrict__ Wout, // (1, 19)
    const float* __restrict__ bout, // (1)
    float* __restrict__ out)        // (SEQ, BATCH, 1)
{
    // GX rows 0..37: full pre-activation for r,z (both biases included);
    // GX rows 38..56: xn + b_in. GH2 row g-32 (g=32..63): h@Whn + b_hn.
    __shared__ float GX[64][17];
    __shared__ float GH2[32][17];

    const int lane  = threadIdx.x;   // 0..31 (wave32)
    const int m     = lane & 15;     // M row within tile
    const int half  = lane >> 4;
    const int bbase = blockIdx.x * 16;

    // ---- B fragments (4x16 f32): VGPR v, lane-half h -> K = 4*ki+2*h+v ----
    // Tiles 0-1 merged over K=40: ki 0..4 = Wih rows, ki 5..9 = Whh rows,
    // with K==19 of the h part carrying b_ih+b_hh (all gates here are r/z).
    v2f Bm[2][2 * NKIT];
    #pragma unroll
    for (int nt = 0; nt < 2; ++nt) {
        const int g = 16 * nt + m;                   // gate column (< 32)
        #pragma unroll
        for (int ki = 0; ki < 2 * NKIT; ++ki) {
            #pragma unroll
            for (int v = 0; v < 2; ++v) {
                float w = 0.0f;
                if (ki < NKIT) {                     // x part
                    const int K = 4 * ki + 2 * half + v;
                    if (K < HID) w = Wih[g * HID + K];
                } else {                             // h part (+ bias row)
                    const int K = 4 * (ki - NKIT) + 2 * half + v;
                    w = (K < HID) ? Whh[g * HID + K] : (bih[g] + bhh[g]);
                }
                Bm[nt][ki][v] = w;
            }
        }
    }
    // Tiles 2-3 split fragments (x and h parts kept separate for the n gate).
    // Bih2 K==19 row is zero; Bhh2 K==19 row: b_ih+b_hh (g<38) or b_hh.
    v2f Bih2[2][NKIT], Bhh2[2][NKIT];
    #pragma unroll
    for (int u = 0; u < 2; ++u) {
        const int g = 32 + 16 * u + m;
        #pragma unroll
        for (int ki = 0; ki < NKIT; ++ki) {
            #pragma unroll
            for (int v = 0; v < 2; ++v) {
                const int K = 4 * ki + 2 * half + v;
                float wi = 0.0f, wh = 0.0f;
                if (g < 57) {
                    if (K < HID) {
                        wi = Wih[g * HID + K];
                        wh = Whh[g * HID + K];
                    } else {
                        wh = (g < 38) ? (bih[g] + bhh[g]) : bhh[g];
                    }
                }
                Bih2[u][ki][v] = wi;
                Bhh2[u][ki][v] = wh;
            }
        }
    }
    // n-gate x-side bias, added at spill time (per-lane scalars).
    const float bn2 = (32 + m >= 38) ? bih[32 + m] : 0.0f;   // tile 2, g=32+m
    const float bn3 = (48 + m < 57)  ? bih[48 + m] : 0.0f;   // tile 3, g=48+m
    const float bo  = bout[0];

    // Output weights for this lane's j set (j = 4*ki + 2*half + v).
    float wo[NKIT][2];
    #pragma unroll
    for (int ki = 0; ki < NKIT; ++ki)
        #pragma unroll
        for (int v = 0; v < 2; ++v) {
            const int j = 4 * ki + 2 * half + v;
            wo[ki][v] = (j < HID) ? Wout[j] : 0.0f;
        }

    // Persistent hidden state in A-layout registers; h0 = 0.
    // Slot K=19 (half1, ki=4, v=1) holds the constant 1.0 bias multiplier.
    v2f ah[NKIT];
    #pragma unroll
    for (int ki = 0; ki < NKIT; ++ki) { ah[ki][0] = 0.0f; ah[ki][1] = 0.0f; }
    ah[NKIT - 1][1] = half ? 1.0f : 0.0f;

    // x A-fragment loader: all loads unconditional (K=19 slot maps to a live
    // but zero-weighted value, so any finite load is fine).
    auto loadA = [&](int t, v2f (&xa)[NKIT]) {
        const float* xp = x + ((size_t)t * BATCH + bbase + m) * HID;
        #pragma unroll
        for (int ki = 0; ki < NKIT - 1; ++ki) {
            const int K0 = 4 * ki + 2 * half;
            xa[ki][0] = xp[K0];
            xa[ki][1] = xp[K0 + 1];
        }
        xa[NKIT - 1][0] = xp[16 + 2 * half];   // K=16 (half0) / K=18 (half1)
        xa[NKIT - 1][1] = xp[17];              // K=17 real; dead for half1
    };

    // Carried x-side accumulators for tiles 2-3 (computed one step ahead).
    v8f dx2 = {}, dx3 = {};

    // One GRU timestep: consumes xc (row t) + carried dx2/dx3, prefetches
    // row t+1 into xn and precomputes its tile-2/3 x-chains.
    auto step = [&](int t, v2f (&xc)[NKIT], v2f (&xn)[NKIT]) {
        loadA((t + 1 < SEQ) ? (t + 1) : (SEQ - 1), xn);

        v8f d0 = {}, d1 = {}, dh2 = {}, dh3 = {};  // inline-0 C operands
        #pragma unroll
        for (int ki = 0; ki < NKIT; ++ki) {        // x half of merged chains
            d0  = WMMA4(xc[ki], Bm[0][ki], d0);    //   + h chains of tiles 2-3
            d1  = WMMA4(xc[ki], Bm[1][ki], d1);
            dh2 = WMMA4(ah[ki], Bhh2[0][ki], dh2);
            dh3 = WMMA4(ah[ki], Bhh2[1][ki], dh3);
        }
        #pragma unroll
        for (int ki = 0; ki < NKIT; ++ki) {        // h half of merged chains
            d0 = WMMA4(ah[ki], Bm[0][NKIT + ki], d0);
            d1 = WMMA4(ah[ki], Bm[1][NKIT + ki], d1);
        }
        #pragma unroll
        for (int r = 0; r < 8; ++r) {
            const int c = r + 8 * half;
            GX[m][c]       = d0[r];                // r,z full pre-activation
            GX[16 + m][c]  = d1[r];
            GX[32 + m][c]  = dx2[r] + ((m < 6) ? dh2[r] : bn2);
            GH2[m][c]      = dh2[r];
            GX[48 + m][c]  = dx3[r] + bn3;
            GH2[16 + m][c] = dh3[r];
        }
        wave_sync();

        // Elementwise gates; lane updates exactly its A-fragment h elements.
        float oacc = 0.0f;
        #pragma unroll
        for (int ki = 0; ki < NKIT; ++ki) {
            #pragma unroll
            for (int v = 0; v < 2; ++v) {
                const int j = 4 * ki + 2 * half + v;           // 0..19
                const float rg = fast_sigmoid(GX[j][m]);
                const float zg = fast_sigmoid(GX[19 + j][m]);
                const float ng = fast_tanh(GX[38 + j][m] + rg * GH2[6 + j][m]);
                const float hn = __builtin_fmaf(zg, ah[ki][v] - ng, ng);
                // Only the (ki=4, v=1) slot can be the K=19 bias slot.
                if (ki == NKIT - 1 && v == 1)
                    ah[ki][v] = half ? 1.0f : hn;
                else
                    ah[ki][v] = hn;
                oacc += hn * wo[ki][v];                        // wo = 0 at j=19
            }
        }

        // Next step's tile-2/3 x-chains: matrix-pipe work that overlaps the
        // transcendental phase above (same barrier region).
        v8f nx2 = {}, nx3 = {};
        #pragma unroll
        for (int ki = 0; ki < NKIT; ++ki) {
            nx2 = WMMA4(xn[ki], Bih2[0][ki], nx2);
            nx3 = WMMA4(xn[ki], Bih2[1][ki], nx3);
        }
        dx2 = nx2;
        dx3 = nx3;

        // Both halves hold the identical sum after the xor-reduce; both store
        // the same value to the same address (no exec divergence).
        oacc += __shfl_xor(oacc, 16, 32);
        __builtin_nontemporal_store(oacc + bo,
                                    &out[(size_t)t * BATCH + bbase + m]);
        wave_sync();
    };

    // Prologue: load row 0 and its tile-2/3 x-chains.
    v2f xbufA[NKIT], xbufB[NKIT];
    loadA(0, xbufA);
    #pragma unroll
    for (int ki = 0; ki < NKIT; ++ki) {
        dx2 = WMMA4(xbufA[ki], Bih2[0][ki], dx2);
        dx3 = WMMA4(xbufA[ki], Bih2[1][ki], dx3);
    }
    wave_sync();

    // Ping-pong x fragments: no register rotation copies.
    for (int t = 0; t < SEQ; t += 2) {
        step(t,     xbufA, xbufB);
        step(t + 1, xbufB, xbufA);
    }
}

extern "C" void kernel_launch(void* const* d_in, const int* in_sizes, int n_in,
                              void* d_out, int out_size, void* d_ws, size_t ws_size,
                              hipStream_t stream) {
    (void)in_sizes; (void)n_in; (void)d_ws; (void)ws_size; (void)out_size;
    const float* x    = (const float*)d_in[0];
    const float* Wih  = (const float*)d_in[1];
    const float* Whh  = (const float*)d_in[2];
    const float* bih  = (const float*)d_in[3];
    const float* bhh  = (const float*)d_in[4];
    const float* Wout = (const float*)d_in[5];
    const float* bout = (const float*)d_in[6];
    float* out = (float*)d_out;

    dim3 grid(BATCH / 16);   // 64 independent scan chains
    dim3 block(32);          // one wave32 each
    gru_wmma_scan<<<grid, block, 0, stream>>>(x, Wih, Whh, bih, bhh, Wout, bout, out);
}